// MulHop_UNetGCN_20151986553228
// MI455X (gfx1250) — compile-verified
//
#include <hip/hip_runtime.h>

// MulHop_UNetGCN for MI455X (gfx1250, wave32).
// Bandwidth-bound workload (edge gather + f32 atomic scatter dominates, ~1.4 GB
// @ 23.3 TB/s); dense GEMMs (~12 GFLOP total) run on the WMMA engine in full
// f32 via V_WMMA_F32_16X16X4_F32 to match the f32 reference exactly.

typedef __attribute__((ext_vector_type(2))) float v2f;
typedef __attribute__((ext_vector_type(8))) float v8f;

#define N_NODES 50000
#define N_EDGES 800000

// ---------------------------------------------------------------------------
// Elementwise / graph kernels
// ---------------------------------------------------------------------------

__global__ void k_deg(const int* __restrict__ dst, float* __restrict__ deg, int E) {
    int i = blockIdx.x * blockDim.x + threadIdx.x;
    if (i < E) atomicAdd(&deg[dst[i]], 1.0f);
}

__global__ void k_dinv(float* __restrict__ d, int n) {
    int i = blockIdx.x * blockDim.x + threadIdx.x;
    if (i < n) d[i] = rsqrtf(d[i] + 2.0f);
}

// One thread per (edge, 4-channel group): float4 gather + 4 atomic f32 adds.
__global__ void k_edge_agg(const int* __restrict__ src, const int* __restrict__ dst,
                           const float* __restrict__ dinv, const float* __restrict__ h,
                           float* __restrict__ agg, int C4, int total) {
    int i = blockIdx.x * blockDim.x + threadIdx.x;
    if (i >= total) return;
    int e  = i / C4;
    int c4 = i - e * C4;
    int s = src[e], d = dst[e];
    float coef = dinv[s] * dinv[d];
    const float4 hv = ((const float4*)h)[(size_t)s * C4 + c4];
    float* ap = agg + ((size_t)d * C4 + c4) * 4;
    atomicAdd(ap + 0, hv.x * coef);
    atomicAdd(ap + 1, hv.y * coef);
    atomicAdd(ap + 2, hv.z * coef);
    atomicAdd(ap + 3, hv.w * coef);
}

// agg[n,c] += 2*dinv[n]^2 * h[n,c] + b[c]   (doubled-self-loop term + bias)
__global__ void k_self_bias(float* __restrict__ agg, const float* __restrict__ h,
                            const float* __restrict__ dinv, const float* __restrict__ b,
                            int C, int total) {
    int i = blockIdx.x * blockDim.x + threadIdx.x;
    if (i >= total) return;
    int n = i / C, c = i - n * C;
    float di = dinv[n];
    agg[i] += 2.0f * di * di * h[i] + b[c];
}

// t1[n, o*2+t] += bd1[o] + s2[n, o]
__global__ void k_t1_epi(float* __restrict__ t1, const float* __restrict__ s2,
                         const float* __restrict__ bd1, int total) {
    int i = blockIdx.x * blockDim.x + threadIdx.x;
    if (i >= total) return;
    int n = i >> 7, j = i & 127;
    int o = j >> 1;
    t1[i] += bd1[o] + s2[(size_t)n * 64 + o];
}

// [N,64,2] (col=o*2+t) -> [N,2,64] (col=t*64+o)
__global__ void k_repack1(float* __restrict__ outp, const float* __restrict__ t1, int total) {
    int i = blockIdx.x * blockDim.x + threadIdx.x;
    if (i >= total) return;
    int n = i >> 7, j = i & 127;
    int t = j >> 6, o = j & 63;
    outp[i] = t1[(size_t)n * 128 + o * 2 + t];
}

// t2p[n, l*32+o] = u[n, t*64 + o*2+tt] + bd2[o] + s1[n,o],  l = 2t+tt
__global__ void k_repack2(float* __restrict__ outp, const float* __restrict__ u,
                          const float* __restrict__ s1, const float* __restrict__ bd2,
                          int total) {
    int i = blockIdx.x * blockDim.x + threadIdx.x;
    if (i >= total) return;
    int n = i >> 7, j = i & 127;
    int l = j >> 5, o = j & 31;
    int t = l >> 1, tt = l & 1;
    outp[i] = u[(size_t)n * 128 + t * 64 + o * 2 + tt] + bd2[o] + s1[(size_t)n * 32 + o];
}

// ---------------------------------------------------------------------------
// Generic f32 WMMA GEMM: C[rows x M] = act(A[rows x K] * B[K x M] + bias)
// One wave per 16x16 output tile; K stepped by 4 (V_WMMA_F32_16X16X4_F32).
// A lane layout (16x4): lanes 0-15 rows 0-15 K={k0,k0+1}; lanes 16-31 K={k0+2,k0+3}.
// B lane layout (4x16): lanes 0-15 cols 0-15 K={k0,k0+1}; lanes 16-31 K={k0+2,k0+3}.
// C/D layout: VGPR r -> row r (lanes 0-15) / row r+8 (lanes 16-31), col = lane&15.
// ---------------------------------------------------------------------------
template <bool RELU>
__global__ void gemm_wmma(const float* __restrict__ A, int lda,
                          const float* __restrict__ B, int M,
                          const float* __restrict__ bias,
                          float* __restrict__ C, int ldc,
                          int nrows, int K) {
    const int lane = threadIdx.x & 31;
    const int wave = (blockIdx.x * blockDim.x + threadIdx.x) >> 5;
    const int tilesM = M >> 4;
    const int tiles  = (nrows >> 4) * tilesM;
    if (wave >= tiles) return;                // uniform per wave: EXEC stays all-ones
    const int tm = wave / tilesM;
    const int tn = wave - tm * tilesM;
    const size_t arow = (size_t)(tm * 16 + (lane & 15)) * (size_t)lda;
    const int col    = tn * 16 + (lane & 15);
    const int khalf  = (lane >> 4) * 2;

    v8f acc = {};
    for (int k0 = 0; k0 < K; k0 += 4) {
        const int ka = k0 + khalf;
        v2f a = *(const v2f*)(A + arow + ka);  // ka even, lda even -> 8B aligned
        v2f b;
        b.x = B[(size_t)ka * M + col];
        b.y = B[(size_t)(ka + 1) * M + col];
        acc = __builtin_amdgcn_wmma_f32_16x16x4_f32(false, a, false, b,
                                                    (short)0, acc, false, false);
    }
    const int rbase = tm * 16 + (lane >> 4) * 8;
    const float bv = bias ? bias[col] : 0.0f;
#pragma unroll
    for (int r = 0; r < 8; ++r) {
        float v = acc[r] + bv;
        if (RELU) v = fmaxf(v, 0.0f);
        C[(size_t)(rbase + r) * (size_t)ldc + col] = v;
    }
}

// ---------------------------------------------------------------------------
// Final GEMM: out[N x 128] = (perm(V) + bd3) @ wf[256 x 128] + bf
// Fuses the [N,4,64] -> [N,256] (o*8 + 2l + tt) permutation + bd3 bias into
// the A-loads, saving a full repack pass over 51 MB.
// ---------------------------------------------------------------------------
__device__ __forceinline__ int permk(int k) {
    int o = k >> 3, L = k & 7;
    return (L >> 1) * 64 + o * 2 + (L & 1);
}

__global__ void gemm_final(const float* __restrict__ V, const float* __restrict__ bd3,
                           const float* __restrict__ wf, const float* __restrict__ bf,
                           float* __restrict__ out, int nrows) {
    const int lane = threadIdx.x & 31;
    const int wave = (blockIdx.x * blockDim.x + threadIdx.x) >> 5;
    const int tilesM = 8;                     // 128 / 16
    const int tiles  = (nrows >> 4) * tilesM;
    if (wave >= tiles) return;
    const int tm = wave / tilesM;
    const int tn = wave - tm * tilesM;
    const float* vrow = V + (size_t)(tm * 16 + (lane & 15)) * 256;
    const int col   = tn * 16 + (lane & 15);
    const int khalf = (lane >> 4) * 2;

    v8f acc = {};
    for (int k0 = 0; k0 < 256; k0 += 4) {
        const int ka = k0 + khalf;
        const int kb = ka + 1;
        v2f a, b;
        a.x = vrow[permk(ka)] + bd3[ka >> 3];
        a.y = vrow[permk(kb)] + bd3[kb >> 3];
        b.x = wf[(size_t)ka * 128 + col];
        b.y = wf[(size_t)kb * 128 + col];
        acc = __builtin_amdgcn_wmma_f32_16x16x4_f32(false, a, false, b,
                                                    (short)0, acc, false, false);
    }
    const int rbase = tm * 16 + (lane >> 4) * 8;
    const float bv = bf[col];
#pragma unroll
    for (int r = 0; r < 8; ++r)
        out[(size_t)(rbase + r) * 128 + col] = acc[r] + bv;
}

// ---------------------------------------------------------------------------
// Host orchestration
// ---------------------------------------------------------------------------
extern "C" void kernel_launch(void* const* d_in, const int* in_sizes, int n_in,
                              void* d_out, int out_size, void* d_ws, size_t ws_size,
                              hipStream_t stream) {
    (void)in_sizes; (void)n_in; (void)out_size; (void)ws_size;

    const float* x       = (const float*)d_in[0];
    const int*   ei      = (const int*)d_in[1];
    const int*   srcp    = ei;            // edge_index[0]
    const int*   dstp    = ei + N_EDGES;  // edge_index[1]
    const float* w_enc1  = (const float*)d_in[2];
    const float* b_enc1  = (const float*)d_in[3];
    const float* fw_enc1 = (const float*)d_in[4];
    const float* fb_enc1 = (const float*)d_in[5];
    const float* w_enc2  = (const float*)d_in[6];
    const float* b_enc2  = (const float*)d_in[7];
    const float* fw_enc2 = (const float*)d_in[8];
    const float* fb_enc2 = (const float*)d_in[9];
    const float* w_enc3  = (const float*)d_in[10];
    const float* b_enc3  = (const float*)d_in[11];
    const float* fw_enc3 = (const float*)d_in[12];
    const float* fb_enc3 = (const float*)d_in[13];
    const float* wd1 = (const float*)d_in[14];  // [128,64,2] -> flat [128x128]
    const float* bd1 = (const float*)d_in[15];
    const float* wd2 = (const float*)d_in[16];  // [64,32,2]  -> flat [64x64]
    const float* bd2 = (const float*)d_in[17];
    const float* wd3 = (const float*)d_in[18];  // [32,32,2]  -> flat [32x64]
    const float* bd3 = (const float*)d_in[19];
    const float* ws1 = (const float*)d_in[20];
    const float* bs1 = (const float*)d_in[21];
    const float* ws2 = (const float*)d_in[22];
    const float* bs2 = (const float*)d_in[23];
    const float* wf  = (const float*)d_in[24];
    const float* bf  = (const float*)d_in[25];
    float* out = (float*)d_out;

    // Workspace carving (~122 MB)
    char* ws = (char*)d_ws;
    size_t off = 0;
    auto carve = [&](size_t bytes) -> float* {
        float* p = (float*)(ws + off);
        off = (off + bytes + 255) & ~(size_t)255;
        return p;
    };
    float* dinv = carve((size_t)N_NODES * 4);
    float* P    = carve((size_t)N_NODES * 128 * 4);  // h / x3 / t1p / t2p
    float* Q    = carve((size_t)N_NODES * 128 * 4);  // agg / s2 / u
    float* X1   = carve((size_t)N_NODES * 32 * 4);
    float* X2   = carve((size_t)N_NODES * 64 * 4);
    float* R    = carve((size_t)N_NODES * 256 * 4);  // t1 / s1 / v

    const int TB = 256;
    auto blocksFor = [](int total) { return (total + 255) / 256; };

    // --- degrees -> dinv = rsqrt(deg + 2) ---
    hipMemsetAsync(dinv, 0, (size_t)N_NODES * 4, stream);
    k_deg <<<blocksFor(N_EDGES), TB, 0, stream>>>(dstp, dinv, N_EDGES);
    k_dinv<<<blocksFor(N_NODES), TB, 0, stream>>>(dinv, N_NODES);

    auto gemm = [&](const float* A, int lda, const float* B, int M, const float* bias,
                    float* C, int ldc, int K, bool relu) {
        int tiles  = (N_NODES / 16) * (M / 16);
        int blocks = (tiles + 7) / 8;            // 8 waves (256 threads) per block
        if (relu)
            gemm_wmma<true ><<<blocks, 256, 0, stream>>>(A, lda, B, M, bias, C, ldc, N_NODES, K);
        else
            gemm_wmma<false><<<blocks, 256, 0, stream>>>(A, lda, B, M, bias, C, ldc, N_NODES, K);
    };

    // --- one multihop GCN layer: Xout = relu((agg(Xin@W)+b) @ fW + fb) ---
    auto layer = [&](const float* Xin, int Cin, const float* W, const float* b,
                     const float* fW, const float* fb, int Cout, float* Xout) {
        gemm(Xin, Cin, W, Cout, nullptr, P, Cout, Cin, false);          // h = Xin @ W
        hipMemsetAsync(Q, 0, (size_t)N_NODES * Cout * 4, stream);
        int C4  = Cout / 4;
        int tot = N_EDGES * C4;
        k_edge_agg<<<blocksFor(tot), TB, 0, stream>>>(srcp, dstp, dinv, P, Q, C4, tot);
        int tot2 = N_NODES * Cout;
        k_self_bias<<<blocksFor(tot2), TB, 0, stream>>>(Q, P, dinv, b, Cout, tot2);
        gemm(Q, Cout, fW, Cout, fb, Xout, Cout, Cout, true);            // relu(.. @ fW + fb)
    };

    layer(x,  128, w_enc1, b_enc1, fw_enc1, fb_enc1, 32,  X1);
    layer(X1, 32,  w_enc2, b_enc2, fw_enc2, fb_enc2, 64,  X2);
    layer(X2, 64,  w_enc3, b_enc3, fw_enc3, fb_enc3, 128, P);           // X3 lives in P
    float* X3 = P;

    // --- decoder ---
    // s2 = X2 @ ws1 + bs1                       -> Q [N,64]
    gemm(X2, 64, ws1, 64, bs1, Q, 64, 64, false);
    // t1 = X3 @ wd1_flat                        -> R [N,128], col = o*2+t
    gemm(X3, 128, wd1, 128, nullptr, R, 128, 128, false);
    // t1 += bd1[o] + s2[n,o]
    k_t1_epi<<<blocksFor(N_NODES * 128), TB, 0, stream>>>(R, Q, bd1, N_NODES * 128);
    // repack [N,64,2] -> [N,2,64]               -> P (X3 dead)
    k_repack1<<<blocksFor(N_NODES * 128), TB, 0, stream>>>(P, R, N_NODES * 128);
    // u[:,t,:] = t1p[:,t,:] @ wd2_flat          -> Q [N,2,64] (s2 dead)
    for (int t = 0; t < 2; ++t)
        gemm(P + t * 64, 128, wd2, 64, nullptr, Q + t * 64, 128, 64, false);
    // s1 = X1 @ ws2 + bs2                       -> R [N,32] (t1 dead)
    gemm(X1, 32, ws2, 32, bs2, R, 32, 32, false);
    // t2p[n, l*32+o] = u + bd2 + s1             -> P (t1p dead)
    k_repack2<<<blocksFor(N_NODES * 128), TB, 0, stream>>>(P, Q, R, bd2, N_NODES * 128);
    // v[:,l,:] = t2p[:,l,:] @ wd3_flat          -> R [N,4,64] (s1 dead)
    for (int l = 0; l < 4; ++l)
        gemm(P + l * 32, 128, wd3, 64, nullptr, R + l * 64, 256, 32, false);
    // out = (perm(v)+bd3) @ wf + bf             -> d_out
    {
        int tiles  = (N_NODES / 16) * 8;
        int blocks = (tiles + 7) / 8;
        gemm_final<<<blocks, 256, 0, stream>>>(R, bd3, wf, bf, out, N_NODES);
    }
}